// SAGEConv_33861522161965
// MI455X (gfx1250) — compile-verified
//
#include <hip/hip_runtime.h>

#define N_NODES 100000
#define N_EDGES 1000000
#define D 64            // D_IN == D_OUT == 64

typedef __attribute__((ext_vector_type(2))) float v2f;
typedef __attribute__((ext_vector_type(8))) float v8f;

// ---------------------------------------------------------------- zero ws
__global__ void sage_zero_kernel(float* __restrict__ ws, int n) {
    int i = blockIdx.x * blockDim.x + threadIdx.x;
    if (i < n) ws[i] = 0.0f;
}

// ---------------------------------------------------------------- degree
__global__ void sage_deg_kernel(const int* __restrict__ dst,
                                float* __restrict__ deg) {
    int e = blockIdx.x * blockDim.x + threadIdx.x;
    if (e < N_EDGES) atomicAdd(&deg[dst[e]], 1.0f);
}

// ---------------------------------------------------------------- scatter-add
// 32 lanes per edge, 2 columns per lane (float2): wave reads one contiguous
// 256B feature row (L2-resident) and issues no-return f32 atomics into nsum.
__global__ void sage_scatter_kernel(const float* __restrict__ feat,
                                    const int* __restrict__ src,
                                    const int* __restrict__ dst,
                                    float* __restrict__ nsum) {
    int t = blockIdx.x * blockDim.x + threadIdx.x;   // 32M threads max
    int e = t >> 5;
    int d = (t & 31) * 2;
    if (e < N_EDGES) {
        int s  = src[e];
        int dn = dst[e];
        const float2 v = *(const float2*)(feat + (size_t)s * D + d);
        atomicAdd(&nsum[(size_t)dn * D + d],     v.x);
        atomicAdd(&nsum[(size_t)dn * D + d + 1], v.y);
    }
}

// ---------------------------------------------------------------- fused GEMM
// One wave per 16x16 output tile. rst = feat@W_self + (nsum/deg)@W_neigh + b.
// Full fp32 via V_WMMA_F32_16X16X4_F32 (K stepped by 4, 32 WMMAs per tile).
__global__ __launch_bounds__(256)
void sage_gemm_kernel(const float* __restrict__ feat,
                      const float* __restrict__ nsum,
                      const float* __restrict__ deg,
                      const float* __restrict__ w_self,
                      const float* __restrict__ b_self,
                      const float* __restrict__ w_neigh,
                      const float* __restrict__ b_neigh,
                      float* __restrict__ out) {
    const int lane = threadIdx.x & 31;
    const int wave = threadIdx.x >> 5;
    const int tile = blockIdx.x * 8 + wave;      // 25000 tiles, exact fit
    const int mt = tile >> 2;                    // 0..6249  (100000/16 rows)
    const int nt = tile & 3;                     // 0..3     (64/16 cols)

    const int lo = lane & 15;                    // M (for A) / N (for B,C,D)
    const int hi = lane >> 4;                    // selects K pair / M+8 half
    const int m  = mt * 16 + lo;                 // this lane's A row
    const int n  = nt * 16 + lo;                 // this lane's B/D column

    v8f acc = {};

    // --- self term: feat @ w_self ---
    const float* arow = feat + (size_t)m * D;
    #pragma unroll
    for (int k0 = 0; k0 < D; k0 += 4) {
        const int k = k0 + 2 * hi;
        v2f a, b;
        a.x = arow[k + 0];
        a.y = arow[k + 1];
        b.x = w_self[(k + 0) * D + n];
        b.y = w_self[(k + 1) * D + n];
        acc = __builtin_amdgcn_wmma_f32_16x16x4_f32(
                  false, a, false, b, (short)0, acc, false, false);
    }

    // --- neighbor term: (nsum * 1/max(deg,1)) @ w_neigh ---
    const float invd = 1.0f / fmaxf(deg[m], 1.0f);
    const float* hrow = nsum + (size_t)m * D;
    #pragma unroll
    for (int k0 = 0; k0 < D; k0 += 4) {
        const int k = k0 + 2 * hi;
        v2f a, b;
        a.x = hrow[k + 0] * invd;
        a.y = hrow[k + 1] * invd;
        b.x = w_neigh[(k + 0) * D + n];
        b.y = w_neigh[(k + 1) * D + n];
        acc = __builtin_amdgcn_wmma_f32_16x16x4_f32(
                  false, a, false, b, (short)0, acc, false, false);
    }

    const float bias = b_self[n] + b_neigh[n];

    // D layout: VGPR r holds M = r (lanes 0-15) / r+8 (lanes 16-31), N = lane%16
    #pragma unroll
    for (int r = 0; r < 8; ++r) {
        const int mm = mt * 16 + r + 8 * hi;
        out[(size_t)mm * D + n] = acc[r] + bias;
    }
}

// ---------------------------------------------------------------- launch
extern "C" void kernel_launch(void* const* d_in, const int* in_sizes, int n_in,
                              void* d_out, int out_size, void* d_ws, size_t ws_size,
                              hipStream_t stream) {
    const float* feat    = (const float*)d_in[0];
    const int*   src     = (const int*)  d_in[1];
    const int*   dst     = (const int*)  d_in[2];
    const float* w_self  = (const float*)d_in[3];
    const float* b_self  = (const float*)d_in[4];
    const float* w_neigh = (const float*)d_in[5];
    const float* b_neigh = (const float*)d_in[6];
    float* out = (float*)d_out;

    float* nsum = (float*)d_ws;                       // [N_NODES * D]
    float* deg  = nsum + (size_t)N_NODES * D;         // [N_NODES]

    // 1) zero accumulators (every call: atomics accumulate otherwise)
    const int nz = N_NODES * D + N_NODES;
    sage_zero_kernel<<<(nz + 255) / 256, 256, 0, stream>>>(nsum, nz);

    // 2) in-degree
    sage_deg_kernel<<<(N_EDGES + 255) / 256, 256, 0, stream>>>(dst, deg);

    // 3) edge scatter-add (32 lanes/edge, float2 per lane)
    const long long st = (long long)N_EDGES * 32;
    sage_scatter_kernel<<<(int)((st + 255) / 256), 256, 0, stream>>>(feat, src, dst, nsum);

    // 4) fused dual-GEMM + mean + bias, one wave per 16x16 tile
    const int tiles  = (N_NODES / 16) * (D / 16);     // 25000
    const int blocks = (tiles + 7) / 8;               // 3125, exact
    sage_gemm_kernel<<<blocks, 256, 0, stream>>>(feat, nsum, deg,
                                                 w_self, b_self,
                                                 w_neigh, b_neigh, out);
}